// RelAttention_22436909155024
// MI455X (gfx1250) — compile-verified
//
#include <hip/hip_runtime.h>
#include <hip/hip_bf16.h>
#include <stdint.h>

#ifndef __has_builtin
#define __has_builtin(x) 0
#endif

// Problem constants (from reference): B=1024, S=64, D=768, REL_TOT=100.
#define SB   64          // sequence length S
#define DD   768         // feature dim D
#define QP   772         // padded LDS row stride in floats (772 % 64 == 4 -> conflict-free
                         // strided column reads for WMMA A-fragments)
#define NTHR 512         // 16 wave32 per block
#define NWAVE 16

typedef __attribute__((ext_vector_type(2))) float v2f;
typedef __attribute__((ext_vector_type(8))) float v8f;
typedef int v4i __attribute__((vector_size(16)));   // matches builtin param type

#if defined(__gfx1250__) && __has_builtin(__builtin_amdgcn_global_load_async_to_lds_b128)
#define HAVE_ASYNC_LDS 1
#else
#define HAVE_ASYNC_LDS 0
#endif

#define AS1 __attribute__((address_space(1)))
#define AS3 __attribute__((address_space(3)))

// Fused relative-attention kernel, one workgroup per batch element b.
//   rel    = relMat[rel_ids[b]]                    (768)
//   w[d]   = W[d] * rel[d]                         (folded scale vector)
//   logits = Q[b] (64x768) . w                     -> 64
//   attn   = softmax(logits)
//   out[b] = attn (1x64) . Q[b]                    -> 768
// Q[b] (192 KiB) is staged once into LDS (async global->LDS, ASYNCcnt),
// both contractions run as V_WMMA_F32_16X16X4_F32 out of LDS.
// Memory-bound workload: 192 MiB of query traffic @ 23.3 TB/s ~= 8.5 us floor;
// this design reads query from HBM exactly once.
__global__ __launch_bounds__(NTHR)
void relattn_fused_kernel(const float* __restrict__ q,       // (B, 64, 768) f32
                          const int*   __restrict__ relIds,  // (B, 1) int
                          const float* __restrict__ W,       // (768,) f32
                          const float* __restrict__ relMat,  // (100, 768) f32
                          float*       __restrict__ out)     // (B, 768) f32
{
    const int b    = blockIdx.x;
    const int tid  = threadIdx.x;
    const int lane = tid & 31;
    const int wid  = tid >> 5;

    __shared__ float qs[SB][QP];        // 64 x 772 floats = 193 KiB (fits 320 KB LDS/WGP)
    __shared__ float wv[DD];            // W * relMat[rid]
    __shared__ float part[4][SB];       // per-k-group partial logits
    __shared__ float attn[SB];
    __shared__ float red[SB];

    const float* __restrict__ qb = q + (size_t)b * (SB * DD);

    // ---- Stage Q[b] into LDS: 64 rows x 192 float4 each = 12288 16B chunks ----
    for (int i = tid; i < SB * (DD / 4); i += NTHR) {
        const int row = i / (DD / 4);
        const int c4  = (i % (DD / 4)) * 4;
        const float* src = qb + row * DD + c4;
        float*       dst = &qs[row][c4];        // 16B aligned: 772*4 = 3088 = 16*193
#if HAVE_ASYNC_LDS
        __builtin_amdgcn_global_load_async_to_lds_b128(
            (AS1 v4i*)(AS1 void*)(void*)src,
            (AS3 v4i*)(AS3 void*)(void*)dst, 0, 0);
#else
        *(float4*)dst = *(const float4*)src;
#endif
    }

    // ---- Folded scale vector w = W * relMat[rid] (overlaps with async copy) ----
    {
        const int rid = relIds[b];              // values in [0,100)
        const float* __restrict__ rrow = relMat + (size_t)rid * DD;
        for (int d = tid; d < DD; d += NTHR) wv[d] = W[d] * rrow[d];
    }

#if HAVE_ASYNC_LDS
#if __has_builtin(__builtin_amdgcn_s_wait_asynccnt)
    __builtin_amdgcn_s_wait_asynccnt(0);
#else
    asm volatile("s_wait_asynccnt 0" ::: "memory");
#endif
#endif
    __syncthreads();

    // =====================================================================
    // Phase 1: logits[s] = sum_d qs[s][d] * wv[d] via V_WMMA_F32_16X16X4_F32
    //   A(16x4): A[m][k] = qs[st*16+m][4c+k]        (M = s within tile)
    //   B(4x16): B[k][n] = wv[4c+k]  (all columns equal -> D columns equal)
    // Wave layout: st = wid&3 selects s-tile, g = wid>>2 selects k-group
    // (48 of 192 four-wide k-chunks each). Partials reduced through LDS.
    // A-frag f32 lane layout: lanes0-15 M=lane {K0,K1}; lanes16-31 M=lane-16 {K2,K3}.
    // =====================================================================
    {
        const int g  = wid >> 2;
        const int st = wid & 3;
        const int m  = lane & 15;
        const int kh = lane >> 4;               // 0: K0/K1, 1: K2/K3
        const float* __restrict__ qrow = &qs[st * 16 + m][0];
        v8f acc = {};
        const int c0 = g * 48;
        for (int c = c0; c < c0 + 48; ++c) {
            const int k = 4 * c + 2 * kh;
            v2f a, bf;
            a.x  = qrow[k];  a.y  = qrow[k + 1];
            bf.x = wv[k];    bf.y = wv[k + 1];
            acc = __builtin_amdgcn_wmma_f32_16x16x4_f32(
                false, a, false, bf, (short)0, acc, false, false);
        }
        // C/D layout: VGPR r -> M=r (lanes 0-15, N=lane) / M=8+r (lanes 16-31).
        // Column N=0 lives in lane 0 (M=0..7) and lane 16 (M=8..15).
        if (lane == 0) {
#pragma unroll
            for (int r = 0; r < 8; ++r) part[g][st * 16 + r] = acc[r];
        }
        if (lane == 16) {
#pragma unroll
            for (int r = 0; r < 8; ++r) part[g][st * 16 + 8 + r] = acc[r];
        }
    }
    __syncthreads();

    // ---- Softmax over s=64 (tiny; LDS tree reduction) ----
    float logit = 0.0f;
    if (tid < SB) {
        logit = part[0][tid] + part[1][tid] + part[2][tid] + part[3][tid];
        red[tid] = logit;
    }
    __syncthreads();
    for (int off = 32; off > 0; off >>= 1) {
        if (tid < off) red[tid] = fmaxf(red[tid], red[tid + off]);
        __syncthreads();
    }
    const float mx = red[0];
    __syncthreads();
    if (tid < SB) {
        const float e = __expf(logit - mx);
        attn[tid] = e;
        red[tid]  = e;
    }
    __syncthreads();
    for (int off = 32; off > 0; off >>= 1) {
        if (tid < off) red[tid] += red[tid + off];
        __syncthreads();
    }
    const float inv = 1.0f / red[0];
    __syncthreads();
    if (tid < SB) attn[tid] *= inv;
    __syncthreads();

    // =====================================================================
    // Phase 2: out[d] = sum_s attn[s] * qs[s][d] via V_WMMA_F32_16X16X4_F32
    //   A(16x4): A[m][k] = attn[4c+k]   (all rows equal -> D rows equal)
    //   B(4x16): B[k][n] = qs[4c+k][d0+n]
    // 48 d-tiles of 16 across 16 waves (3 each); K accumulates the s=64 dim.
    // =====================================================================
    {
        const int kh = lane >> 4;
        const int nn = lane & 15;
        for (int dt = wid; dt < 48; dt += NWAVE) {
            const int d0 = dt * 16;
            v8f acc = {};
#pragma unroll
            for (int c = 0; c < 16; ++c) {
                const int k = 4 * c + 2 * kh;   // s index
                v2f a, bf;
                a.x  = attn[k];          a.y  = attn[k + 1];
                bf.x = qs[k][d0 + nn];   bf.y = qs[k + 1][d0 + nn];
                acc = __builtin_amdgcn_wmma_f32_16x16x4_f32(
                    false, a, false, bf, (short)0, acc, false, false);
            }
            // All D rows equal; row M=0 sits in VGPR0 of lanes 0-15 (N=lane).
            if (lane < 16) out[(size_t)b * DD + d0 + nn] = acc[0];
        }
    }
}

extern "C" void kernel_launch(void* const* d_in, const int* in_sizes, int n_in,
                              void* d_out, int out_size, void* d_ws, size_t ws_size,
                              hipStream_t stream) {
    (void)n_in; (void)d_ws; (void)ws_size; (void)out_size;
    const float* q      = (const float*)d_in[0];   // (B, 64, 768) f32
    const int*   relIds = (const int*)d_in[1];     // (B, 1) int (harness transfers ints as i32)
    const float* W      = (const float*)d_in[2];   // (768,) f32
    const float* relMat = (const float*)d_in[3];   // (100, 768) f32
    float*       out    = (float*)d_out;           // (B, 768) f32

    const int B = in_sizes[1];                     // rel_ids has B elements
    relattn_fused_kernel<<<dim3(B), dim3(NTHR), 0, stream>>>(q, relIds, W, relMat, out);
}